// BPNNP_9560597200960
// MI455X (gfx1250) — compile-verified
//
#include <hip/hip_runtime.h>
#include <math.h>

#define A_ATOMS   8192
#define NNEI      24
#define NF        8
#define WAVES_PB  8
#define PI_F      3.14159265358979323846f

typedef __attribute__((ext_vector_type(2))) float v2f;
typedef __attribute__((ext_vector_type(8))) float v8f;

__global__ __launch_bounds__(256)
void bpnn_sf_kernel(const float* __restrict__ n_diff,
                    const float* __restrict__ n_dist,
                    const int*   __restrict__ j_elems,
                    const float* __restrict__ eta2,
                    const float* __restrict__ R_s,
                    const float* __restrict__ R_c2,
                    const float* __restrict__ zeta,
                    const float* __restrict__ Lambda,
                    const float* __restrict__ eta4,
                    const float* __restrict__ R_c4,
                    float* __restrict__ out)
{
    __shared__ float s_dist[WAVES_PB][32];
    __shared__ float s_inv [WAVES_PB][32];           // 1/r  (0 for padding lanes)
    __shared__ float s_dx  [WAVES_PB][32];
    __shared__ float s_dy  [WAVES_PB][32];
    __shared__ float s_dz  [WAVES_PB][32];
    __shared__ float s_m1  [WAVES_PB][32];
    __shared__ float s_m8  [WAVES_PB][32];
    __shared__ float s_fc4 [WAVES_PB][32][NF];       // fc(r, Rc4[f])
    __shared__ float s_e4  [WAVES_PB][32][NF];       // exp(-eta4[f] * r^2)
    __shared__ float s_out [WAVES_PB][32];

    const int lid  = threadIdx.x & 31;   // lane within wave (wave32)
    const int wid  = threadIdx.x >> 5;   // wave within block
    const int atom = blockIdx.x * WAVES_PB + wid;

    // ---- per-feature parameters (uniform; compiler scalarizes) ----
    float eta2v[NF], Rsv[NF], piRc2[NF], zetav[NF], lamv[NF], eta4v[NF], piRc4[NF], pwv[NF];
#pragma unroll
    for (int f = 0; f < NF; ++f) {
        eta2v[f] = eta2[f];          Rsv[f]   = R_s[f];
        piRc2[f] = PI_F / R_c2[f];   piRc4[f] = PI_F / R_c4[f];
        zetav[f] = zeta[f];          lamv[f]  = Lambda[f];
        eta4v[f] = eta4[f];          pwv[f]   = exp2f(1.0f - zetav[f]);
    }
    const float Rc40 = R_c4[0];      // cutoff check uses R_c4[0] (as in reference)

    // ---- lane n owns neighbor n of this atom (n >= NNEI are zero padding) ----
    const int  n  = lid;
    const bool vn = (n < NNEI);
    const long long p = (long long)atom * NNEI + n;

    float d = 0.f, dx = 0.f, dy = 0.f, dz = 0.f, m1 = 0.f, m8 = 0.f;
    if (vn) {
        d  = n_dist[p];
        dx = n_diff[3 * p + 0];
        dy = n_diff[3 * p + 1];
        dz = n_diff[3 * p + 2];
        const int e = j_elems[p];
        m1 = (e == 1) ? 1.f : 0.f;
        m8 = (e == 8) ? 1.f : 0.f;
    }
    s_dist[wid][lid] = d;
    s_inv [wid][lid] = (d > 0.f) ? (1.0f / d) : 0.f;  // 24 divides/atom instead of 576
    s_dx[wid][lid] = dx;  s_dy[wid][lid] = dy;  s_dz[wid][lid] = dz;
    s_m1[wid][lid] = m1;  s_m8[wid][lid] = m8;
    {
        const float d2 = d * d;
#pragma unroll
        for (int f = 0; f < NF; ++f) {
            s_fc4[wid][lid][f] = vn ? fmaf(cosf(d * piRc4[f]), 0.5f, 0.5f) : 0.f;
            s_e4 [wid][lid][f] = vn ? expf(-eta4v[f] * d2) : 0.f;  // separable half of p2
        }
    }

    __syncthreads();

    // acc[0:8]=g2_H, [8:16]=g2_O, [16:24]=g4_HO, [24:32]=g4_HH
    float acc[32];
#pragma unroll
    for (int i = 0; i < 32; ++i) acc[i] = 0.f;

    // ---- G2 (each lane contributes its one neighbor) ----
    if (vn) {
#pragma unroll
        for (int f = 0; f < NF; ++f) {
            const float dd = d - Rsv[f];
            const float sf = expf(-eta2v[f] * dd * dd) * fmaf(cosf(d * piRc2[f]), 0.5f, 0.5f);
            acc[f]     = sf * m1;
            acc[8 + f] = sf * m8;
        }
    }

    // ---- Gram matrix dot[j][k] = diff_j . diff_k via f32 WMMA 16x16x4 ----
    // A 16x4 layout: lanes 0-15 VGPR{0,1} = (K0,K1) of row M=lane; lanes 16-31 = (K2,K3).
    // B 4x16 uses the symmetric transpose striping, so for a Gram matrix the
    // A-fragment of row-block `b` equals the B-fragment of col-block `b`.
    const bool lo = (lid < 16);
    const int  ml = lid & 15;
    v2f f0, f16;
    f0[0]  = lo ? s_dx[wid][ml]      : s_dz[wid][ml];
    f0[1]  = lo ? s_dy[wid][ml]      : 0.f;           // K=3 padding
    f16[0] = lo ? s_dx[wid][16 + ml] : s_dz[wid][16 + ml];
    f16[1] = lo ? s_dy[wid][16 + ml] : 0.f;

    v8f T00, T01, T10, T11;
#if __has_builtin(__builtin_amdgcn_wmma_f32_16x16x4_f32)
    {
        v8f c0 = {};
        T00 = __builtin_amdgcn_wmma_f32_16x16x4_f32(false, f0,  false, f0,  (short)0, c0, false, false);
        T01 = __builtin_amdgcn_wmma_f32_16x16x4_f32(false, f0,  false, f16, (short)0, c0, false, false);
        T10 = __builtin_amdgcn_wmma_f32_16x16x4_f32(false, f16, false, f0,  (short)0, c0, false, false);
        T11 = __builtin_amdgcn_wmma_f32_16x16x4_f32(false, f16, false, f16, (short)0, c0, false, false);
    }
#else
    // Scalar fallback: same C-matrix register layout, computed from LDS.
    {
        const int h8 = (lid >> 4) * 8;
#pragma unroll
        for (int r = 0; r < 8; ++r) {
            const int j0 = r + h8, j1 = 16 + r + h8;
            const int k0 = ml,     k1 = 16 + ml;
            T00[r] = s_dx[wid][j0]*s_dx[wid][k0] + s_dy[wid][j0]*s_dy[wid][k0] + s_dz[wid][j0]*s_dz[wid][k0];
            T01[r] = s_dx[wid][j0]*s_dx[wid][k1] + s_dy[wid][j0]*s_dy[wid][k1] + s_dz[wid][j0]*s_dz[wid][k1];
            T10[r] = s_dx[wid][j1]*s_dx[wid][k0] + s_dy[wid][j1]*s_dy[wid][k0] + s_dz[wid][j1]*s_dz[wid][k0];
            T11[r] = s_dx[wid][j1]*s_dx[wid][k1] + s_dy[wid][j1]*s_dy[wid][k1] + s_dz[wid][j1]*s_dz[wid][k1];
        }
    }
#endif

    // ---- G4: walk the C-matrix tiles. VGPR r of a 16x16 f32 tile holds
    // (M = r, N = lane) for lanes 0-15 and (M = r+8, N = lane-16) for lanes 16-31.
    const int hi8 = (lid >> 4) * 8;
    auto processTile = [&](int tj, int tk, const v8f& D) {
        const int kk = tk * 16 + ml;
        if (kk >= NNEI) return;
        const float rk   = s_dist[wid][kk];
        const float invk = s_inv [wid][kk];
        const float m1k  = s_m1  [wid][kk];
        const float m8k  = s_m8  [wid][kk];
#pragma unroll
        for (int r = 0; r < 8; ++r) {
            const int jj = tj * 16 + r + hi8;
            if (jj >= NNEI) continue;
            const float m1j = s_m1[wid][jj];
            const float w18 = m1j * m8k;                          // H-O pairs
            const float w11 = (jj < kk) ? (m1j * m1k) : 0.f;      // H-H triu(k=1)
            if (w18 == 0.f && w11 == 0.f) continue;
            const float dotv = D[r];
            const float rj   = s_dist[wid][jj];
            const float sq   = rj * rj + rk * rk - 2.f * dotv;
            const float djk  = (sq > 0.f) ? sqrtf(sq) : 0.f;
            if (!(djk < Rc40)) continue;                          // valid &= d_jk < R_c[0]
            const float cosv = dotv * s_inv[wid][jj] * invk;
#pragma unroll
            for (int f = 0; f < NF; ++f) {
                // (1 + lam*cos)^zeta : base >= 0 mathematically; clamp rounding
                // noise so exp2(zeta*log2(base)) is valid (log2(0)=-inf -> 0).
                const float base = fmaxf(fmaf(cosv, lamv[f], 1.0f), 0.0f);
                const float p1   = exp2f(zetav[f] * log2f(base));
                const float p2   = s_e4[wid][jj][f] * s_e4[wid][kk][f];  // exp(-eta*(rj^2+rk^2))
                const float fcd  = fmaf(cosf(djk * piRc4[f]), 0.5f, 0.5f);
                const float p3   = s_fc4[wid][jj][f] * s_fc4[wid][kk][f] * fcd;
                const float t    = p1 * p2 * p3 * pwv[f];
                acc[16 + f] += t * w18;
                acc[24 + f] += t * w11;
            }
        }
    };
    processTile(0, 0, T00);
    processTile(0, 1, T01);
    processTile(1, 0, T10);
    processTile(1, 1, T11);

    // ---- wave-wide butterfly reduction of all 32 accumulators ----
#pragma unroll
    for (int i = 0; i < 32; ++i) {
        float v = acc[i];
#pragma unroll
        for (int off = 16; off > 0; off >>= 1)
            v += __shfl_xor(v, off, 32);
        acc[i] = v;
    }
    if (lid == 0) {
#pragma unroll
        for (int i = 0; i < 32; ++i) s_out[wid][i] = acc[i];
    }
    __syncthreads();
    out[(long long)atom * 32 + lid] = s_out[wid][lid];   // coalesced row store
}

extern "C" void kernel_launch(void* const* d_in, const int* in_sizes, int n_in,
                              void* d_out, int out_size, void* d_ws, size_t ws_size,
                              hipStream_t stream) {
    const float* n_diff = (const float*)d_in[0];
    const float* n_dist = (const float*)d_in[1];
    // d_in[2] = atom_i_idx: unused — the neighbor list is dense (atom = p / NNEI).
    const int*   j_elems = (const int*)d_in[3];
    const float* eta2   = (const float*)d_in[4];
    const float* R_s    = (const float*)d_in[5];
    const float* R_c2   = (const float*)d_in[6];
    const float* zeta   = (const float*)d_in[7];
    const float* Lambda = (const float*)d_in[8];
    const float* eta4   = (const float*)d_in[9];
    const float* R_c4   = (const float*)d_in[10];
    float* out = (float*)d_out;

    dim3 grid(A_ATOMS / WAVES_PB);
    dim3 block(256);
    hipLaunchKernelGGL(bpnn_sf_kernel, grid, block, 0, stream,
                       n_diff, n_dist, j_elems,
                       eta2, R_s, R_c2, zeta, Lambda, eta4, R_c4, out);
}